// StructureModule_4337916970073
// MI455X (gfx1250) — compile-verified
//
#include <hip/hip_runtime.h>
#include <math.h>

// ---------------------------------------------------------------------------
// AF2 structure module for MI455X (gfx1250), fp32 via V_WMMA_F32_16X16X4_F32.
// B=2, L=1024, D=128, H=8, C=16, PQ=PV=4, ITERS=4.
// ---------------------------------------------------------------------------

typedef __attribute__((ext_vector_type(2))) float v2f;
typedef __attribute__((ext_vector_type(8))) float v8f;

#define BB    2
#define LL    1024
#define DD    128
#define HH    8
#define CC    16
#define NROWS (BB * LL)      // 2048
#define FEATD 256            // H*C + H*PV*3 + H*PV
#define HIDD  512            // transition hidden

// fused projection panel: [wq|wk|wv|wqp|wkp|wvp] -> 672 columns
#define PSTR  672
#define QOFF  0
#define KOFF  128
#define VOFF  256
#define QPOFF 384
#define KPOFF 480
#define VPOFF 576

__device__ __forceinline__ v8f wmma4(v2f a, v2f b, v8f c) {
  // D(16x16,f32) += A(16x4,f32) * B(4x16,f32)
  return __builtin_amdgcn_wmma_f32_16x16x4_f32(false, a, false, b, (short)0, c,
                                               false, false);
}
__device__ __forceinline__ v8f zero8() { v8f z = {}; return z; }

__device__ __forceinline__ float gelu_erf(float x) {
  return 0.5f * x * (1.0f + erff(x * 0.70710678118654752f));
}

// ---------------------------------------------------------------------------
// LayerNorm: one wave per row of D=128.
// ---------------------------------------------------------------------------
__global__ void ln_kernel(const float* __restrict__ in, float* __restrict__ out,
                          const float* __restrict__ g, const float* __restrict__ b) {
  int row  = blockIdx.x;
  int lane = threadIdx.x;             // 0..31
  const float4 v = *(const float4*)(in + (size_t)row * DD + lane * 4);
  float s  = v.x + v.y + v.z + v.w;
  float ss = v.x * v.x + v.y * v.y + v.z * v.z + v.w * v.w;
  for (int m = 1; m < 32; m <<= 1) {
    s  += __shfl_xor(s, m, 32);
    ss += __shfl_xor(ss, m, 32);
  }
  float mean = s * (1.0f / DD);
  float var  = ss * (1.0f / DD) - mean * mean;
  float rstd = rsqrtf(var + 1e-5f);
  int c = lane * 4;
  float4 o;
  o.x = (v.x - mean) * rstd * g[c + 0] + b[c + 0];
  o.y = (v.y - mean) * rstd * g[c + 1] + b[c + 1];
  o.z = (v.z - mean) * rstd * g[c + 2] + b[c + 2];
  o.w = (v.w - mean) * rstd * g[c + 3] + b[c + 3];
  *(float4*)(out + (size_t)row * DD + c) = o;
}

// ---------------------------------------------------------------------------
// Concatenate projection weights into one 128 x 672 panel (once per call).
// ---------------------------------------------------------------------------
__global__ void concat_w_kernel(const float* __restrict__ wq, const float* __restrict__ wk,
                                const float* __restrict__ wv, const float* __restrict__ wqp,
                                const float* __restrict__ wkp, const float* __restrict__ wvp,
                                float* __restrict__ wcat) {
  int idx = blockIdx.x * blockDim.x + threadIdx.x;
  if (idx >= DD * PSTR) return;
  int k = idx / PSTR, c = idx % PSTR;
  float v;
  if      (c < KOFF)  v = wq[k * DD + c];
  else if (c < VOFF)  v = wk[k * DD + (c - KOFF)];
  else if (c < QPOFF) v = wv[k * DD + (c - VOFF)];
  else if (c < KPOFF) v = wqp[k * 96 + (c - QPOFF)];
  else if (c < VPOFF) v = wkp[k * 96 + (c - KPOFF)];
  else                v = wvp[k * 96 + (c - VPOFF)];
  wcat[idx] = v;
}

// ---------------------------------------------------------------------------
// LDS-staged WMMA GEMM: out[N x M] = act(A[N x K] @ W[K x M] + bias) + resid
// One block = 8 waves sharing a 16-row A tile staged in LDS (padded stride).
// Each wave iterates over 16-col tiles; B panels stream from global with
// prefetch of the next tile.  K % 4 == 0, M % 16 == 0.
// ---------------------------------------------------------------------------
__global__ void gemm_lds_kernel(const float* __restrict__ A, const float* __restrict__ W,
                                const float* __restrict__ bias,
                                const float* __restrict__ resid,
                                float* __restrict__ out, int K, int M, int gelu_flag) {
  extern __shared__ float Atile[];    // 16 * (K+4) floats
  int row0 = blockIdx.x * 16;
  int tid  = threadIdx.x;             // 0..255
  int lane = tid & 31;
  int wave = tid >> 5;                // 0..7
  int Ks   = K + 4;

  // cooperative stage of the 16 x K A tile (float4, coalesced)
  for (int idx = tid * 4; idx < 16 * K; idx += 256 * 4) {
    int r = idx / K, c = idx % K;     // K % 4 == 0 -> c % 4 == 0
    float4 vv = *(const float4*)(A + (size_t)(row0 + r) * K + c);
    *(float4*)(Atile + r * Ks + c) = vv;
  }
  __syncthreads();

  int ml = lane & 15;
  int hl = lane >> 4;
  int ntiles = M / 16;
  const float* arow = Atile + ml * Ks;

  for (int ct = wave; ct < ntiles; ct += 8) {
    int col0 = ct * 16;
    if (ct + 8 < ntiles) __builtin_prefetch(W + (ct + 8) * 16, 0, 1);
    v8f acc = zero8();
    for (int kk = 0; kk < K; kk += 4) {
      v2f a, bf;
      a.x  = arow[kk + 2 * hl];
      a.y  = arow[kk + 2 * hl + 1];
      bf.x = W[(size_t)(kk + 2 * hl) * M + col0 + ml];
      bf.y = W[(size_t)(kk + 2 * hl + 1) * M + col0 + ml];
      acc = wmma4(a, bf, acc);
    }
    int col = col0 + ml;
    float bv = bias ? bias[col] : 0.0f;
#pragma unroll
    for (int i = 0; i < 8; i++) {
      int row = row0 + i + 8 * hl;
      float v = acc[i] + bv;
      if (gelu_flag) v = gelu_erf(v);
      if (resid) v += resid[(size_t)row * M + col];
      out[(size_t)row * M + col] = v;
    }
  }
}

// ---------------------------------------------------------------------------
// Frame transform: qg/kg/vg = R @ p + t, plus |qg|^2, |kg|^2 sums over points.
// Reads points from fused projection output (stride PSTR).
// ---------------------------------------------------------------------------
__global__ void frame_kernel(const float* __restrict__ pout,
                             const float* __restrict__ R, const float* __restrict__ t,
                             float* __restrict__ qg, float* __restrict__ kg,
                             float* __restrict__ vg,
                             float* __restrict__ q2, float* __restrict__ k2) {
  int n = blockIdx.x * blockDim.x + threadIdx.x;
  if (n >= NROWS * HH) return;
  int row = n / HH, hh = n % HH;
  int b = row / LL, l = row % LL;
  float Rm[9], tv[3];
#pragma unroll
  for (int j = 0; j < 9; j++) Rm[j] = R[(size_t)row * 9 + j];
#pragma unroll
  for (int j = 0; j < 3; j++) tv[j] = t[(size_t)row * 3 + j];

  const float* base = pout + (size_t)row * PSTR;
  float q2s = 0.0f, k2s = 0.0f;
#pragma unroll
  for (int p = 0; p < 4; p++) {
    const float* sq = base + QPOFF + hh * 12 + p * 3;
    float gx = Rm[0] * sq[0] + Rm[1] * sq[1] + Rm[2] * sq[2] + tv[0];
    float gy = Rm[3] * sq[0] + Rm[4] * sq[1] + Rm[5] * sq[2] + tv[1];
    float gz = Rm[6] * sq[0] + Rm[7] * sq[1] + Rm[8] * sq[2] + tv[2];
    float* dq = qg + ((size_t)row * HH + hh) * 12 + p * 3;
    dq[0] = gx; dq[1] = gy; dq[2] = gz;
    q2s += gx * gx + gy * gy + gz * gz;

    const float* sk = base + KPOFF + hh * 12 + p * 3;
    gx = Rm[0] * sk[0] + Rm[1] * sk[1] + Rm[2] * sk[2] + tv[0];
    gy = Rm[3] * sk[0] + Rm[4] * sk[1] + Rm[5] * sk[2] + tv[1];
    gz = Rm[6] * sk[0] + Rm[7] * sk[1] + Rm[8] * sk[2] + tv[2];
    float* dk = kg + ((size_t)row * HH + hh) * 12 + p * 3;
    dk[0] = gx; dk[1] = gy; dk[2] = gz;
    k2s += gx * gx + gy * gy + gz * gz;

    const float* sv = base + VPOFF + hh * 12 + p * 3;
    gx = Rm[0] * sv[0] + Rm[1] * sv[1] + Rm[2] * sv[2] + tv[0];
    gy = Rm[3] * sv[0] + Rm[4] * sv[1] + Rm[5] * sv[2] + tv[1];
    gz = Rm[6] * sv[0] + Rm[7] * sv[1] + Rm[8] * sv[2] + tv[2];
    float* dv = vg + ((size_t)row * HH + hh) * 12 + p * 3;
    dv[0] = gx; dv[1] = gy; dv[2] = gz;
  }
  q2[((size_t)(b * HH + hh)) * LL + l] = q2s;
  k2[((size_t)(b * HH + hh)) * LL + l] = k2s;
}

// ---------------------------------------------------------------------------
// IPA attention: one wave per (b, head, 16-query tile).  ONE-PASS online
// softmax: logits via WMMA (q.kT + point cross term), per-chunk row max via
// half-wave butterflies, accumulators rescaled in C-layout registers, a@v and
// a@vg accumulated via WMMA after an LDS C->A layout round trip.
// ---------------------------------------------------------------------------
__global__ void attn_kernel(const float* __restrict__ pout,
                            const float* __restrict__ qg, const float* __restrict__ kg,
                            const float* __restrict__ vg,
                            const float* __restrict__ q2, const float* __restrict__ k2,
                            const float* __restrict__ gamma_p,
                            const unsigned char* __restrict__ mask,
                            float* __restrict__ o_out, float* __restrict__ opg_out) {
  __shared__ float lds_p[256];
  __shared__ float lds_q2[16];
  int l0 = blockIdx.x * 16;
  int hh = blockIdx.y;
  int b  = blockIdx.z;
  int lane = threadIdx.x;
  int ml = lane & 15;
  int hl = lane >> 4;
  const size_t rowQ = (size_t)b * LL + l0;

  float g0 = gamma_p[hh];
  float sp = (g0 > 20.0f) ? g0 : log1pf(expf(g0));
  const float WLc    = 0.70710678118654752f;            // sqrt(0.5)
  const float pointw = WLc * sp * 0.11785113019775793f; // WL*gamma*WC*0.5
  const float sscale = WLc * 0.25f;                     // WL / sqrt(C)

  // A-layout fragments for q (16x16) and qg (16x12)
  v2f qa[4], qga[3];
  const float* qrow = pout + (rowQ + ml) * PSTR + QOFF + hh * CC;
#pragma unroll
  for (int kc = 0; kc < 4; kc++) {
    qa[kc].x = qrow[kc * 4 + 2 * hl];
    qa[kc].y = qrow[kc * 4 + 2 * hl + 1];
  }
  const float* qgrow = qg + ((rowQ + ml) * HH + hh) * 12;
#pragma unroll
  for (int kc = 0; kc < 3; kc++) {
    qga[kc].x = qgrow[kc * 4 + 2 * hl];
    qga[kc].y = qgrow[kc * 4 + 2 * hl + 1];
  }
  if (lane < 16) lds_q2[lane] = q2[((size_t)(b * HH + hh)) * LL + l0 + lane];
  __syncthreads();
  float q2r[8];
#pragma unroll
  for (int i = 0; i < 8; i++) q2r[i] = lds_q2[i + 8 * hl];

  v8f o_acc = zero8(), op_acc = zero8();
  float rm[8], rs[8];
#pragma unroll
  for (int i = 0; i < 8; i++) { rm[i] = -3.0e38f; rs[i] = 0.0f; }

  for (int m0 = 0; m0 < LL; m0 += 16) {
    // ---- logits tile (C layout) ----
    v8f accs = zero8(), accp = zero8();
    const float* krow = pout + ((size_t)b * LL + m0 + ml) * PSTR + KOFF + hh * CC;
#pragma unroll
    for (int kc = 0; kc < 4; kc++) {
      v2f bf;
      bf.x = krow[kc * 4 + 2 * hl];
      bf.y = krow[kc * 4 + 2 * hl + 1];
      accs = wmma4(qa[kc], bf, accs);
    }
    const float* kgrow = kg + (((size_t)b * LL + m0 + ml) * HH + hh) * 12;
#pragma unroll
    for (int kc = 0; kc < 3; kc++) {
      v2f bf;
      bf.x = kgrow[kc * 4 + 2 * hl];
      bf.y = kgrow[kc * 4 + 2 * hl + 1];
      accp = wmma4(qga[kc], bf, accp);
    }
    float k2c = k2[((size_t)(b * HH + hh)) * LL + m0 + ml];
    bool valid = mask[(size_t)b * LL + m0 + ml] != 0;

    // ---- online softmax update ----
#pragma unroll
    for (int i = 0; i < 8; i++) {
      float d2 = q2r[i] + k2c - 2.0f * accp[i];
      float lg = sscale * accs[i] - pointw * d2;
      lg = valid ? lg : -1e9f;
      float cm = lg;
      for (int m = 1; m < 16; m <<= 1) cm = fmaxf(cm, __shfl_xor(cm, m, 32));
      float nm = fmaxf(rm[i], cm);
      float sc = expf(rm[i] - nm);
      rm[i] = nm;
      float p = expf(lg - nm);
      rs[i] = rs[i] * sc + p;
      o_acc[i]  *= sc;
      op_acc[i] *= sc;
      lds_p[(i + 8 * hl) * 16 + ml] = p;  // C-layout -> LDS row-major tile
    }
    __syncthreads();
    v2f pa[4];
#pragma unroll
    for (int kc = 0; kc < 4; kc++) {      // LDS -> A-layout fragments
      pa[kc].x = lds_p[ml * 16 + kc * 4 + 2 * hl];
      pa[kc].y = lds_p[ml * 16 + kc * 4 + 2 * hl + 1];
    }
#pragma unroll
    for (int kc = 0; kc < 4; kc++) {
      int kr = m0 + kc * 4 + 2 * hl;
      v2f vb;
      vb.x = pout[((size_t)b * LL + kr) * PSTR + VOFF + hh * CC + ml];
      vb.y = pout[((size_t)b * LL + kr + 1) * PSTR + VOFF + hh * CC + ml];
      o_acc = wmma4(pa[kc], vb, o_acc);
      v2f gb;
      gb.x = (ml < 12) ? vg[(((size_t)b * LL + kr) * HH + hh) * 12 + ml] : 0.0f;
      gb.y = (ml < 12) ? vg[(((size_t)b * LL + kr + 1) * HH + hh) * 12 + ml] : 0.0f;
      op_acc = wmma4(pa[kc], gb, op_acc);
    }
    __syncthreads();
  }

#pragma unroll
  for (int i = 0; i < 8; i++) {
    for (int m = 1; m < 16; m <<= 1) rs[i] += __shfl_xor(rs[i], m, 32);
    float inv = 1.0f / rs[i];
    size_t row = rowQ + i + 8 * hl;
    o_out[row * DD + hh * CC + ml] = o_acc[i] * inv;
    if (ml < 12) opg_out[(row * HH + hh) * 12 + ml] = op_acc[i] * inv;
  }
}

// ---------------------------------------------------------------------------
// Assemble feat = [o | R^T(op_g - t) | norms]; one thread per (row, head).
// ---------------------------------------------------------------------------
__global__ void finalize_kernel(const float* __restrict__ o, const float* __restrict__ opg,
                                const float* __restrict__ R, const float* __restrict__ t,
                                float* __restrict__ feat) {
  int n = blockIdx.x * blockDim.x + threadIdx.x;
  if (n >= NROWS * HH) return;
  int row = n / HH, hh = n % HH;
#pragma unroll
  for (int c = 0; c < CC; c++)
    feat[(size_t)row * FEATD + hh * CC + c] = o[(size_t)row * DD + hh * CC + c];
  float Rm[9], tv[3];
#pragma unroll
  for (int j = 0; j < 9; j++) Rm[j] = R[(size_t)row * 9 + j];
#pragma unroll
  for (int j = 0; j < 3; j++) tv[j] = t[(size_t)row * 3 + j];
#pragma unroll
  for (int p = 0; p < 4; p++) {
    const float* gp = opg + ((size_t)row * HH + hh) * 12 + p * 3;
    float gx = gp[0] - tv[0], gy = gp[1] - tv[1], gz = gp[2] - tv[2];
    float l0 = Rm[0] * gx + Rm[3] * gy + Rm[6] * gz;  // R^T
    float l1 = Rm[1] * gx + Rm[4] * gy + Rm[7] * gz;
    float l2 = Rm[2] * gx + Rm[5] * gy + Rm[8] * gz;
    float* fd = feat + (size_t)row * FEATD + 128 + hh * 12 + p * 3;
    fd[0] = l0; fd[1] = l1; fd[2] = l2;
    feat[(size_t)row * FEATD + 224 + hh * 4 + p] =
        sqrtf(l0 * l0 + l1 * l1 + l2 * l2 + 1e-8f);
  }
}

// ---------------------------------------------------------------------------
// Backbone update: u = h2 @ w2 + b2; Rodrigues; R = rot@R; t += R@u[3:6].
// ---------------------------------------------------------------------------
__global__ void bb_update_kernel(const float* __restrict__ h2, const float* __restrict__ w2,
                                 const float* __restrict__ b2,
                                 float* __restrict__ R, float* __restrict__ t) {
  int row = blockIdx.x * blockDim.x + threadIdx.x;
  if (row >= NROWS) return;
  float u[6];
#pragma unroll
  for (int j = 0; j < 6; j++) u[j] = b2[j];
  const float* hr = h2 + (size_t)row * DD;
  for (int c = 0; c < DD; c++) {
    float hv = hr[c];
    const float* wr = w2 + c * 6;
#pragma unroll
    for (int j = 0; j < 6; j++) u[j] += hv * wr[j];
  }
  float th = sqrtf(u[0] * u[0] + u[1] * u[1] + u[2] * u[2]);
  th = fmaxf(th, 1e-8f);
  float inv = 1.0f / th;
  float ax = u[0] * inv, ay = u[1] * inv, az = u[2] * inv;
  float ct = cosf(th), st = sinf(th), oc = 1.0f - ct;
  float rot[9];
  rot[0] = ct + oc * ax * ax;      rot[1] = oc * ax * ay - st * az; rot[2] = oc * ax * az + st * ay;
  rot[3] = oc * ay * ax + st * az; rot[4] = ct + oc * ay * ay;      rot[5] = oc * ay * az - st * ax;
  rot[6] = oc * az * ax - st * ay; rot[7] = oc * az * ay + st * ax; rot[8] = ct + oc * az * az;
  float Ro[9], Rn[9];
#pragma unroll
  for (int j = 0; j < 9; j++) Ro[j] = R[(size_t)row * 9 + j];
#pragma unroll
  for (int i = 0; i < 3; i++)
#pragma unroll
    for (int j = 0; j < 3; j++)
      Rn[i * 3 + j] = rot[i * 3 + 0] * Ro[0 * 3 + j] + rot[i * 3 + 1] * Ro[1 * 3 + j] +
                      rot[i * 3 + 2] * Ro[2 * 3 + j];
#pragma unroll
  for (int j = 0; j < 9; j++) R[(size_t)row * 9 + j] = Rn[j];
#pragma unroll
  for (int i = 0; i < 3; i++)
    t[(size_t)row * 3 + i] += Rn[i * 3 + 0] * u[3] + Rn[i * 3 + 1] * u[4] + Rn[i * 3 + 2] * u[5];
}

__global__ void init_frames_kernel(float* __restrict__ R, float* __restrict__ t) {
  int row = blockIdx.x * blockDim.x + threadIdx.x;
  if (row >= NROWS) return;
#pragma unroll
  for (int i = 0; i < 3; i++) {
#pragma unroll
    for (int j = 0; j < 3; j++) R[(size_t)row * 9 + i * 3 + j] = (i == j) ? 1.0f : 0.0f;
    t[(size_t)row * 3 + i] = 0.0f;
  }
}

__global__ void coords_kernel(const float* __restrict__ R, const float* __restrict__ t,
                              float* __restrict__ out) {
  int row = blockIdx.x * blockDim.x + threadIdx.x;
  if (row >= NROWS) return;
  const float ideal[3][3] = {{-0.527f, 1.359f, 0.0f}, {0.0f, 0.0f, 0.0f}, {1.526f, 0.0f, 0.0f}};
  float Rm[9], tv[3];
#pragma unroll
  for (int j = 0; j < 9; j++) Rm[j] = R[(size_t)row * 9 + j];
#pragma unroll
  for (int j = 0; j < 3; j++) tv[j] = t[(size_t)row * 3 + j];
#pragma unroll
  for (int a = 0; a < 3; a++)
#pragma unroll
    for (int i = 0; i < 3; i++)
      out[(size_t)row * 9 + a * 3 + i] =
          Rm[i * 3 + 0] * ideal[a][0] + Rm[i * 3 + 1] * ideal[a][1] +
          Rm[i * 3 + 2] * ideal[a][2] + tv[i];
#pragma unroll
  for (int j = 0; j < 9; j++) out[NROWS * 9 + (size_t)row * 9 + j] = Rm[j];
#pragma unroll
  for (int i = 0; i < 3; i++) out[NROWS * 18 + (size_t)row * 3 + i] = tv[i];
}

// ---------------------------------------------------------------------------
extern "C" void kernel_launch(void* const* d_in, const int* in_sizes, int n_in,
                              void* d_out, int out_size, void* d_ws, size_t ws_size,
                              hipStream_t stream) {
  (void)in_sizes; (void)n_in; (void)out_size; (void)ws_size;
  const float* s_in = (const float*)d_in[0];
  const unsigned char* mask = (const unsigned char*)d_in[1];
  const float* ipa_ln_s = (const float*)d_in[2];
  const float* ipa_ln_b = (const float*)d_in[3];
  const float* wq  = (const float*)d_in[4];
  const float* wk  = (const float*)d_in[5];
  const float* wv  = (const float*)d_in[6];
  const float* wqp = (const float*)d_in[7];
  const float* wkp = (const float*)d_in[8];
  const float* wvp = (const float*)d_in[9];
  const float* gamma = (const float*)d_in[10];
  const float* wo  = (const float*)d_in[11];
  const float* bo  = (const float*)d_in[12];
  const float* t_ln_s = (const float*)d_in[13];
  const float* t_ln_b = (const float*)d_in[14];
  const float* t_w1 = (const float*)d_in[15];
  const float* t_b1 = (const float*)d_in[16];
  const float* t_w2 = (const float*)d_in[17];
  const float* t_b2 = (const float*)d_in[18];
  const float* bb_ln_s = (const float*)d_in[19];
  const float* bb_ln_b = (const float*)d_in[20];
  const float* bb_w1 = (const float*)d_in[21];
  const float* bb_b1 = (const float*)d_in[22];
  const float* bb_w2 = (const float*)d_in[23];
  const float* bb_b2 = (const float*)d_in[24];

  float* w = (float*)d_ws;
  size_t off = 0;
  float* S    = w + off; off += (size_t)NROWS * DD;     // state
  float* X    = w + off; off += (size_t)NROWS * DD;     // LN output
  float* POUT = w + off; off += (size_t)NROWS * PSTR;   // fused q|k|v|qp|kp|vp
  float* QG   = w + off; off += (size_t)NROWS * 96;
  float* KG   = w + off; off += (size_t)NROWS * 96;
  float* VG   = w + off; off += (size_t)NROWS * 96;
  float* Q2   = w + off; off += (size_t)NROWS * HH;
  float* K2   = w + off; off += (size_t)NROWS * HH;
  float* O    = w + off; off += (size_t)NROWS * DD;
  float* OPG  = w + off; off += (size_t)NROWS * 96;
  float* FEAT = w + off; off += (size_t)NROWS * FEATD;
  float* H1   = w + off; off += (size_t)NROWS * HIDD;
  float* H2   = w + off; off += (size_t)NROWS * DD;
  float* WCAT = w + off; off += (size_t)DD * PSTR;
  float* Rb   = w + off; off += (size_t)NROWS * 9;
  float* Tb   = w + off; off += (size_t)NROWS * 3;

  hipMemcpyAsync(S, s_in, (size_t)NROWS * DD * sizeof(float),
                 hipMemcpyDeviceToDevice, stream);
  init_frames_kernel<<<(NROWS + 255) / 256, 256, 0, stream>>>(Rb, Tb);
  concat_w_kernel<<<(DD * PSTR + 255) / 256, 256, 0, stream>>>(
      wq, wk, wv, wqp, wkp, wvp, WCAT);

  const int gemm_blocks = NROWS / 16;   // 128
#define SH(K) ((16 * ((K) + 4)) * sizeof(float))

  for (int it = 0; it < 4; ++it) {
    // ---- IPA ----
    ln_kernel<<<NROWS, 32, 0, stream>>>(S, X, ipa_ln_s, ipa_ln_b);
    gemm_lds_kernel<<<gemm_blocks, 256, SH(DD), stream>>>(
        X, WCAT, nullptr, nullptr, POUT, DD, PSTR, 0);
    frame_kernel<<<(NROWS * HH + 255) / 256, 256, 0, stream>>>(
        POUT, Rb, Tb, QG, KG, VG, Q2, K2);
    attn_kernel<<<dim3(LL / 16, HH, BB), 32, 0, stream>>>(
        POUT, QG, KG, VG, Q2, K2, gamma, mask, O, OPG);
    finalize_kernel<<<(NROWS * HH + 255) / 256, 256, 0, stream>>>(O, OPG, Rb, Tb, FEAT);
    gemm_lds_kernel<<<gemm_blocks, 256, SH(FEATD), stream>>>(
        FEAT, wo, bo, S, S, FEATD, DD, 0);                       // residual

    // ---- Transition MLP ----
    ln_kernel<<<NROWS, 32, 0, stream>>>(S, X, t_ln_s, t_ln_b);
    gemm_lds_kernel<<<gemm_blocks, 256, SH(DD), stream>>>(
        X, t_w1, t_b1, nullptr, H1, DD, HIDD, 1);
    gemm_lds_kernel<<<gemm_blocks, 256, SH(HIDD), stream>>>(
        H1, t_w2, t_b2, S, S, HIDD, DD, 0);                      // residual

    // ---- Backbone update ----
    ln_kernel<<<NROWS, 32, 0, stream>>>(S, X, bb_ln_s, bb_ln_b);
    gemm_lds_kernel<<<gemm_blocks, 256, SH(DD), stream>>>(
        X, bb_w1, bb_b1, nullptr, H2, DD, DD, 1);
    bb_update_kernel<<<(NROWS + 255) / 256, 256, 0, stream>>>(H2, bb_w2, bb_b2, Rb, Tb);
  }

  coords_kernel<<<(NROWS + 255) / 256, 256, 0, stream>>>(Rb, Tb, (float*)d_out);
#undef SH
}